// SESConv2dFT_67765993996487
// MI455X (gfx1250) — compile-verified
//
#include <hip/hip_runtime.h>
#include <math.h>

typedef __attribute__((ext_vector_type(2))) float v2f;
typedef __attribute__((ext_vector_type(8))) float v8f;

#define NBATCH 32
#define CH     256
#define NSP    32
#define NF     17
#define BINS   (NSP * NF)       // 544
#define PI2    6.28318530717958647692f

__device__ __forceinline__ v8f wmma4(v2f a, v2f b, v8f c) {
  // D = A(16x4 f32) * B(4x16 f32) + C(16x16 f32)
  return __builtin_amdgcn_wmma_f32_16x16x4_f32(false, a, false, b, (short)0, c,
                                               false, false);
}

__global__ void k_zero(float* acc) {
  if (threadIdx.x < 8) acc[threadIdx.x] = 0.0f;
}

// H[bin][p][c] = e^{-i*2pi*(u+v)/32} * sum_{i,j<3} w[p][c][i][j] e^{+i*2pi*(u*i+v*j)/32}
__global__ void k_hfft(const float* __restrict__ w, float* __restrict__ Hr,
                       float* __restrict__ Hi) {
  __shared__ float twc[32], tws[32];
  if (threadIdx.x < 32) {
    float a = PI2 * (float)threadIdx.x * (1.0f / 32.0f);
    twc[threadIdx.x] = cosf(a);
    tws[threadIdx.x] = sinf(a);
  }
  __syncthreads();
  int p = blockIdx.x & 255;
  int bin = blockIdx.x >> 8;
  int u = bin / NF, v = bin % NF;
  int c = threadIdx.x;
  const float* wp = w + ((size_t)p * CH + c) * 9;
  float sr = 0.f, si = 0.f;
#pragma unroll
  for (int i = 0; i < 3; ++i)
#pragma unroll
    for (int j = 0; j < 3; ++j) {
      int m = (u * i + v * j) & 31;
      float ww = wp[i * 3 + j];
      sr += ww * twc[m];
      si += ww * tws[m];
    }
  int ph = (u + v) & 31;
  float cr = twc[ph], sn = tws[ph];
  size_t o = ((size_t)bin * CH + p) * CH + c;
  Hr[o] = cr * sr + sn * si;
  Hi[o] = cr * si - sn * sr;
}

// T[img][a][v] = sum_b x[img][a][b] e^{-i*2pi*v*b/32}
__global__ void k_dft_rows(const float* __restrict__ x, float* __restrict__ Tr,
                           float* __restrict__ Ti) {
  __shared__ float twc[32], tws[32];
  if (threadIdx.x < 32) {
    float a = PI2 * (float)threadIdx.x * (1.0f / 32.0f);
    twc[threadIdx.x] = cosf(a);
    tws[threadIdx.x] = sinf(a);
  }
  __syncthreads();
  int idx = blockIdx.x * blockDim.x + threadIdx.x;   // exact grid
  int v = idx % NF;
  int rest = idx / NF;
  int a = rest & 31;
  int img = rest >> 5;
  const float* row = x + ((size_t)img * 32 + a) * 32;
  float sr = 0.f, si = 0.f;
  for (int b = 0; b < 32; ++b) {
    int m = (v * b) & 31;
    float ww = row[b];
    sr += ww * twc[m];
    si -= ww * tws[m];
  }
  Tr[idx] = sr;
  Ti[idx] = si;
}

// X[bin=u*17+v][c][bb] = sum_a T[bb*256+c][a][v] e^{-i*2pi*u*a/32}
// Stored pair-interleaved for b64 WMMA B-fragment loads:
//   Xp[((bin*128 + c/2)*32 + bb)*2 + (c&1)]
__global__ void k_dft_cols(const float* __restrict__ Tr, const float* __restrict__ Ti,
                           float* __restrict__ Xr, float* __restrict__ Xi) {
  __shared__ float twc[32], tws[32];
  if (threadIdx.x < 32) {
    float a = PI2 * (float)threadIdx.x * (1.0f / 32.0f);
    twc[threadIdx.x] = cosf(a);
    tws[threadIdx.x] = sinf(a);
  }
  __syncthreads();
  int idx = blockIdx.x * blockDim.x + threadIdx.x;
  int bb = idx & 31;
  int c = (idx >> 5) & 255;
  int bin = idx >> 13;
  int u = bin / NF, v = bin % NF;
  size_t base = (size_t)(bb * CH + c) * (32 * NF) + v;
  float sr = 0.f, si = 0.f;
  for (int a = 0; a < 32; ++a) {
    int m = (u * a) & 31;
    float tr = Tr[base + (size_t)a * NF];
    float ti = Ti[base + (size_t)a * NF];
    sr += tr * twc[m] + ti * tws[m];
    si += ti * twc[m] - tr * tws[m];
  }
  size_t o = ((size_t)(bin * (CH / 2) + (c >> 1)) * NBATCH + bb) * 2 + (c & 1);
  Xr[o] = sr;
  Xi[o] = si;
}

// Per-bin complex GEMM: Y = H @ X, f32 WMMA, 32(M) x 32(N) register tile per wave.
// 3 accumulators per 16x16 sub-tile: Yr = yrp - yrm, Yi = chained (HrXi + HiXr).
__global__ void __launch_bounds__(256, 1)
k_ygemm(const float* __restrict__ Hr, const float* __restrict__ Hi,
        const float* __restrict__ Xr, const float* __restrict__ Xi,
        float* __restrict__ Yr, float* __restrict__ Yi) {
  int wid = (blockIdx.x * blockDim.x + threadIdx.x) >> 5;  // 544*8 waves exact
  int lane = threadIdx.x & 31;
  int bin = wid >> 3;
  int p0 = (wid & 7) * 32;
  int r = lane & 15;
  int half = lane >> 4;

  const float* HrB = Hr + ((size_t)bin * CH + (p0 + r)) * CH + 2 * half;
  const float* HiB = Hi + ((size_t)bin * CH + (p0 + r)) * CH + 2 * half;
  // pair layout: elem = bin*8192 + (k/2 + half)*64 + col*2
  const float* XrB = Xr + (size_t)bin * CH * NBATCH + (size_t)half * 64 + 2 * r;
  const float* XiB = Xi + (size_t)bin * CH * NBATCH + (size_t)half * 64 + 2 * r;

  v8f yrp[2][2], yrm[2][2], yim[2][2];
#pragma unroll
  for (int mi = 0; mi < 2; ++mi)
#pragma unroll
    for (int ni = 0; ni < 2; ++ni) {
      yrp[mi][ni] = (v8f){};
      yrm[mi][ni] = (v8f){};
      yim[mi][ni] = (v8f){};
    }

  for (int k = 0; k < CH; k += 4) {
    v2f ar[2], ai[2], br[2], bi[2];
#pragma unroll
    for (int mi = 0; mi < 2; ++mi) {
      ar[mi] = *(const v2f*)(HrB + (size_t)mi * 16 * CH + k);
      ai[mi] = *(const v2f*)(HiB + (size_t)mi * 16 * CH + k);
    }
#pragma unroll
    for (int ni = 0; ni < 2; ++ni) {
      br[ni] = *(const v2f*)(XrB + (size_t)k * 32 + ni * 32);
      bi[ni] = *(const v2f*)(XiB + (size_t)k * 32 + ni * 32);
    }
#pragma unroll
    for (int mi = 0; mi < 2; ++mi)
#pragma unroll
      for (int ni = 0; ni < 2; ++ni) {
        yrp[mi][ni] = wmma4(ar[mi], br[ni], yrp[mi][ni]);
        yrm[mi][ni] = wmma4(ai[mi], bi[ni], yrm[mi][ni]);
        yim[mi][ni] = wmma4(ar[mi], bi[ni], yim[mi][ni]);
        yim[mi][ni] = wmma4(ai[mi], br[ni], yim[mi][ni]);
      }
  }
  size_t base = (size_t)bin * CH * NBATCH;
#pragma unroll
  for (int mi = 0; mi < 2; ++mi)
#pragma unroll
    for (int ni = 0; ni < 2; ++ni) {
      int col = ni * 16 + r;
#pragma unroll
      for (int e = 0; e < 8; ++e) {
        int row = p0 + mi * 16 + e + 8 * half;
        Yr[base + (size_t)row * NBATCH + col] = yrp[mi][ni][e] - yrm[mi][ni][e];
        Yi[base + (size_t)row * NBATCH + col] = yim[mi][ni][e];
      }
    }
}

// Upper-triangular 32x32 block-pairs of HHT[bin] = conj(H) H^T ; sum |HHT|^2 (p<q).
// 3 accumulators per sub-tile: Re chained (HrHr' + HiHi'), Im = imp - imm.
__global__ void __launch_bounds__(256, 1)
k_hht_ang(const float* __restrict__ Hr, const float* __restrict__ Hi,
          float* __restrict__ angAcc) {
  int wid = (blockIdx.x * blockDim.x + threadIdx.x) >> 5;  // 544*36 waves exact
  int lane = threadIdx.x & 31;
  int bin = wid / 36;
  int tt = wid % 36;
  int i = 0;
  while (tt >= 8 - i) { tt -= 8 - i; ++i; }
  int j = i + tt;
  int p0 = i * 32, q0 = j * 32;
  int r = lane & 15;
  int half = lane >> 4;

  const float* ArB = Hr + ((size_t)bin * CH + (p0 + r)) * CH + 2 * half;
  const float* AiB = Hi + ((size_t)bin * CH + (p0 + r)) * CH + 2 * half;
  const float* BrB = Hr + ((size_t)bin * CH + (q0 + r)) * CH + 2 * half;  // B = H^T tile
  const float* BiB = Hi + ((size_t)bin * CH + (q0 + r)) * CH + 2 * half;

  v8f re[2][2], imp[2][2], imm[2][2];
#pragma unroll
  for (int mi = 0; mi < 2; ++mi)
#pragma unroll
    for (int ni = 0; ni < 2; ++ni) {
      re[mi][ni] = (v8f){};
      imp[mi][ni] = (v8f){};
      imm[mi][ni] = (v8f){};
    }

  for (int k = 0; k < CH; k += 4) {
    v2f ar[2], ai[2], br[2], bi[2];
#pragma unroll
    for (int mi = 0; mi < 2; ++mi) {
      ar[mi] = *(const v2f*)(ArB + (size_t)mi * 16 * CH + k);
      ai[mi] = *(const v2f*)(AiB + (size_t)mi * 16 * CH + k);
      br[mi] = *(const v2f*)(BrB + (size_t)mi * 16 * CH + k);
      bi[mi] = *(const v2f*)(BiB + (size_t)mi * 16 * CH + k);
    }
#pragma unroll
    for (int mi = 0; mi < 2; ++mi)
#pragma unroll
      for (int ni = 0; ni < 2; ++ni) {
        re[mi][ni] = wmma4(ar[mi], br[ni], re[mi][ni]);
        re[mi][ni] = wmma4(ai[mi], bi[ni], re[mi][ni]);
        imp[mi][ni] = wmma4(ar[mi], bi[ni], imp[mi][ni]);
        imm[mi][ni] = wmma4(ai[mi], br[ni], imm[mi][ni]);
      }
  }
  float partial = 0.f;
#pragma unroll
  for (int mi = 0; mi < 2; ++mi)
#pragma unroll
    for (int ni = 0; ni < 2; ++ni) {
      int q = q0 + ni * 16 + r;
#pragma unroll
      for (int e = 0; e < 8; ++e) {
        int p = p0 + mi * 16 + e + 8 * half;
        float rev = re[mi][ni][e];
        float imv = imp[mi][ni][e] - imm[mi][ni][e];
        if (q > p) partial += rev * rev + imv * imv;
      }
    }
  for (int off = 16; off > 0; off >>= 1) partial += __shfl_xor(partial, off, 32);
  if (lane == 0) atomicAdd(angAcc, partial);
}

// Row/Col norm losses: sum (sqrt(sum |H|^2) - 1)^2 into acc[0] (row) / acc[1] (col).
__global__ void k_norm(const float* __restrict__ Hr, const float* __restrict__ Hi,
                       float* __restrict__ acc, int isRow) {
  int idx = blockIdx.x * blockDim.x + threadIdx.x;  // 544*256 exact
  int bin = idx >> 8;
  int a = idx & 255;
  size_t base;
  int stride;
  if (isRow) { base = ((size_t)bin * CH + a) * CH; stride = 1; }
  else       { base = (size_t)bin * CH * CH + a;   stride = CH; }
  float s = 0.f;
  for (int k = 0; k < CH; ++k) {
    float hr = Hr[base + (size_t)k * stride];
    float hi = Hi[base + (size_t)k * stride];
    s += hr * hr + hi * hi;
  }
  float d = sqrtf(s) - 1.0f;
  float part = d * d;
  for (int off = 16; off > 0; off >>= 1) part += __shfl_xor(part, off, 32);
  if ((threadIdx.x & 31) == 0) atomicAdd(&acc[isRow ? 0 : 1], part);
}

// t[bb][p][a][v] = sum_u Y[u*17+v][p][bb] e^{+i*2pi*u*a/32}
__global__ void k_idft_u(const float* __restrict__ Yr, const float* __restrict__ Yi,
                         float* __restrict__ Tr, float* __restrict__ Ti) {
  __shared__ float twc[32], tws[32];
  if (threadIdx.x < 32) {
    float a = PI2 * (float)threadIdx.x * (1.0f / 32.0f);
    twc[threadIdx.x] = cosf(a);
    tws[threadIdx.x] = sinf(a);
  }
  __syncthreads();
  int idx = blockIdx.x * blockDim.x + threadIdx.x;
  int v = idx % NF;
  int rest = idx / NF;
  int a = rest & 31;
  int p = (rest >> 5) & 255;
  int bb = rest >> 13;
  float sr = 0.f, si = 0.f;
  for (int u = 0; u < 32; ++u) {
    int m = (u * a) & 31;
    size_t yo = ((size_t)(u * NF + v) * CH + p) * NBATCH + bb;
    float yr = Yr[yo];
    float yi = Yi[yo];
    sr += yr * twc[m] - yi * tws[m];
    si += yr * tws[m] + yi * twc[m];
  }
  Tr[idx] = sr;
  Ti[idx] = si;
}

// y[bb][p][a][b] = (1/1024)*(t[..,0] + 2*sum_{v=1..15} Re{t e^{+i2pi v b/32}} + (-1)^b t[..,16]) + bias
__global__ void k_idft_v(const float* __restrict__ Tr, const float* __restrict__ Ti,
                         const float* __restrict__ bias, float* __restrict__ out) {
  __shared__ float twc[32], tws[32];
  if (threadIdx.x < 32) {
    float a = PI2 * (float)threadIdx.x * (1.0f / 32.0f);
    twc[threadIdx.x] = cosf(a);
    tws[threadIdx.x] = sinf(a);
  }
  __syncthreads();
  int idx = blockIdx.x * blockDim.x + threadIdx.x;  // 8388608 exact
  int b = idx & 31;
  int a = (idx >> 5) & 31;
  int p = (idx >> 10) & 255;
  int bb = idx >> 18;
  size_t base = (size_t)(((bb * CH + p) * 32) + a) * NF;
  float acc = Tr[base];  // v = 0
#pragma unroll
  for (int v = 1; v < 16; ++v) {
    int m = (v * b) & 31;
    acc += 2.0f * (Tr[base + v] * twc[m] - Ti[base + v] * tws[m]);
  }
  acc += ((b & 1) ? -1.0f : 1.0f) * Tr[base + 16];
  out[idx] = acc * (1.0f / 1024.0f) + bias[p];
}

__global__ void k_finalize(const float* __restrict__ acc, float* __restrict__ outL) {
  float rowLoss = acc[0] * (1.0f / (float)(BINS * CH));
  float colLoss = acc[1] * (1.0f / (float)(BINS * CH));
  float angLoss = acc[2] * (1.0f / 17756160.0f);  // bins * cout*(cout-1)/2
  *outL = 256.0f * colLoss + 256.0f * rowLoss + angLoss;
}

extern "C" void kernel_launch(void* const* d_in, const int* in_sizes, int n_in,
                              void* d_out, int out_size, void* d_ws, size_t ws_size,
                              hipStream_t stream) {
  const float* x = (const float*)d_in[0];     // (32,256,32,32)
  const float* w = (const float*)d_in[1];     // (256,256,3,3)
  const float* bias = (const float*)d_in[2];  // (256,)
  float* out = (float*)d_out;                 // y (8388608) + L (1)
  float* ws = (float*)d_ws;

  const size_t nH = (size_t)BINS * CH * CH;       // 35,651,584
  const size_t nX = (size_t)BINS * CH * NBATCH;   //  4,456,448
  float* Hr = ws;
  float* Hi = Hr + nH;
  float* Xr = Hi + nH;
  float* Xi = Xr + nX;
  float* Tr = Xi + nX;
  float* Ti = Tr + nX;
  float* Yr = Ti + nX;
  float* Yi = Yr + nX;
  float* acc = Yi + nX;  // [0]=rowAcc [1]=colAcc [2]=angAcc

  hipLaunchKernelGGL(k_zero, dim3(1), dim3(32), 0, stream, acc);
  hipLaunchKernelGGL(k_hfft, dim3(BINS * CH), dim3(CH), 0, stream, w, Hr, Hi);
  hipLaunchKernelGGL(k_dft_rows, dim3(17408), dim3(256), 0, stream, x, Tr, Ti);
  hipLaunchKernelGGL(k_dft_cols, dim3(17408), dim3(256), 0, stream, Tr, Ti, Xr, Xi);
  hipLaunchKernelGGL(k_ygemm, dim3(544), dim3(256), 0, stream, Hr, Hi, Xr, Xi, Yr, Yi);
  hipLaunchKernelGGL(k_hht_ang, dim3(2448), dim3(256), 0, stream, Hr, Hi, acc + 2);
  hipLaunchKernelGGL(k_norm, dim3(544), dim3(256), 0, stream, Hr, Hi, acc, 1);
  hipLaunchKernelGGL(k_norm, dim3(544), dim3(256), 0, stream, Hr, Hi, acc, 0);
  hipLaunchKernelGGL(k_idft_u, dim3(17408), dim3(256), 0, stream, Yr, Yi, Tr, Ti);
  hipLaunchKernelGGL(k_idft_v, dim3(32768), dim3(256), 0, stream, Tr, Ti, bias, out);
  hipLaunchKernelGGL(k_finalize, dim3(1), dim3(1), 0, stream, acc, out + 8388608);
}